// GFlowLoss_53077205844108
// MI455X (gfx1250) — compile-verified
//
#include <hip/hip_runtime.h>
#include <math.h>

// Problem constants (from the reference)
#define LRc   0.1f
#define B1c   0.9f
#define B2c   0.999f
#define EPSc  1e-8f
#define OEPSc 0.1f
#define TOLc  1e-3f
// NM=240, NT=240, N=256, 4*NM=960, 2*NM=480, NM*N=61440, 61440/2=30720

typedef __attribute__((ext_vector_type(2))) float v2f;
typedef __attribute__((ext_vector_type(8))) float v8f;

// Workspace layout (float offsets). Total ~346,628 floats ~= 1.39 MB.
enum : int {
  OFF_GL   = 0,       // 240*240 G_latent
  OFF_MG   = 57600,   // Adam m for G
  OFF_VG   = 115200,  // Adam v for G
  OFF_S    = 172800,  // sigmoid(G_latent)
  OFF_Q    = 230400,  // q[u,w] = dL/dP * P
  OFF_GG   = 288000,  // grad wrt G_latent
  OFF_TAU  = 345600,  // tau (256)
  OFF_MT   = 345856,
  OFF_VT   = 346112,
  OFF_GT   = 346368,  // grad tau
  OFF_LOSS = 346624,  // [0]=L_odd, [1]=L_order
  OFF_ACT  = 346626   // int: while-loop "still running" flag
};

__global__ void k_init(const float* __restrict__ gl0,
                       const float* __restrict__ tau0, float* ws) {
  int idx = blockIdx.x * blockDim.x + threadIdx.x;
  if (idx < 57600) { ws[OFF_GL + idx] = gl0[idx]; ws[OFF_MG + idx] = 0.f; ws[OFF_VG + idx] = 0.f; }
  if (idx < 256)   { ws[OFF_TAU + idx] = tau0[idx]; ws[OFF_MT + idx] = 0.f; ws[OFF_VT + idx] = 0.f; }
  if (idx == 0)    { ((int*)(ws + OFF_ACT))[0] = 1; }
}

__global__ void k_sigmoid(float* ws, int finalPass) {
  if (!finalPass && ((const int*)(ws + OFF_ACT))[0] == 0) return;
  int idx = blockIdx.x * blockDim.x + threadIdx.x;
  if (idx < 2) ws[OFF_LOSS + idx] = 0.f;           // reset loss accumulators
  if (idx < 57600) {
    float x = ws[OFF_GL + idx];
    ws[OFF_S + idx] = 1.f / (1.f + __expf(-x));
  }
}

// P[u,w] = prod_c (1 - 2*A[w,c+16]*S[u,c]);  q = (P-T)/480 * P;  L_odd += (P-T)^2/960
__global__ void k_forward(const float* __restrict__ A, float* ws, int finalPass) {
  if (!finalPass && ((const int*)(ws + OFF_ACT))[0] == 0) return;
  __shared__ float sS[16 * 241];   // 241 stride: conflict-free under wave32 lane map
  __shared__ float sA2[16 * 241];
  __shared__ float sRed[256];
  const int tid = threadIdx.x, tx = tid & 15, ty = tid >> 4;
  const int u0 = blockIdx.y * 16, w0 = blockIdx.x * 16;
  for (int i = tid; i < 16 * 240; i += 256) {
    int r = i / 240, c = i - r * 240;
    sS[r * 241 + c]  = ws[OFF_S + (u0 + r) * 240 + c];
    sA2[r * 241 + c] = -2.f * A[(w0 + r) * 256 + c + 16];
  }
  __syncthreads();
  const float* ps = &sS[ty * 241];
  const float* pa = &sA2[tx * 241];
  float p = 1.f;
#pragma unroll 4
  for (int c = 0; c < 240; ++c) p *= fmaf(pa[c], ps[c], 1.f);
  const int u = u0 + ty, w = w0 + tx;
  const float T = (u == w) ? -1.f : 1.f;
  const float d = p - T;
  ws[OFF_Q + u * 240 + w] = d * (1.f / 480.f) * p;
  sRed[tid] = d * d * (1.f / 960.f);
  __syncthreads();
  if (tid < 32) {  // wave 0, EXEC all-ones (uniform branch): WMMA-as-reduction
    v8f acc = {0.f, 0.f, 0.f, 0.f, 0.f, 0.f, 0.f, 0.f};
    v2f one = {1.f, 1.f};
#pragma unroll
    for (int ch = 0; ch < 4; ++ch) {
      v2f b;
      b.x = sRed[ch * 64 + tid * 2];
      b.y = sRed[ch * 64 + tid * 2 + 1];
      // D = ones(16x4) x B(4x16) + D : folds 64 partials/instruction, all rows equal
      acc = __builtin_amdgcn_wmma_f32_16x16x4_f32(false, one, false, b,
                                                  (short)0, acc, false, false);
    }
    float r = acc[0];
#pragma unroll
    for (int m = 16; m; m >>= 1) r += __shfl_xor(r, m, 32);
    if (tid == 0) atomicAdd(ws + OFF_LOSS + 0, 0.5f * r);  // rows duplicated -> /2
  }
}

// grad_Gl[u,c] = (sum_w q[u,w]*a2[w,c]/(1+a2*S) + relu(diff)^2/61440) * S*(1-S)
__global__ void k_gradG(const float* __restrict__ A, float* ws) {
  if (((const int*)(ws + OFF_ACT))[0] == 0) return;
  __shared__ float sQ[16 * 241];
  __shared__ float sA2c[240 * 16];
  const int tid = threadIdx.x, tx = tid & 15, ty = tid >> 4;
  const int u0 = blockIdx.y * 16, c0 = blockIdx.x * 16;
  for (int i = tid; i < 16 * 240; i += 256) {
    int r = i / 240, w = i - r * 240;
    sQ[r * 241 + w] = ws[OFF_Q + (u0 + r) * 240 + w];
  }
  for (int i = tid; i < 240 * 16; i += 256) {
    int w = i >> 4, cc = i & 15;
    sA2c[i] = -2.f * A[w * 256 + c0 + cc + 16];
  }
  __syncthreads();
  const int u = u0 + ty, c = c0 + tx;
  const float s = ws[OFF_S + u * 240 + c];
  const float* pq = &sQ[ty * 241];
  float acc = 0.f;
#pragma unroll 4
  for (int w = 0; w < 240; ++w) {
    float a2 = sA2c[w * 16 + tx];
    float t = fmaf(a2, s, 1.f);
    acc = fmaf(pq[w] * a2, __builtin_amdgcn_rcpf(t), acc);
  }
  float dd = ws[OFF_TAU + u] - ws[OFF_TAU + c + 16] + OEPSc;
  float rl = fmaxf(dd, 0.f);
  float gS = acc + rl * rl * (1.f / 61440.f);
  ws[OFF_GG + u * 240 + c] = gS * s * (1.f - s);
}

// L_order and grad_tau. One block per tau index t.
__global__ void k_order(float* ws, int finalPass) {
  if (!finalPass && ((const int*)(ws + OFF_ACT))[0] == 0) return;
  __shared__ float r0[256], r1[256], r2[256];
  const int t = blockIdx.x, tid = threadIdx.x;
  const float taut = ws[OFF_TAU + t];
  float rowg = 0.f, rowl = 0.f, colg = 0.f;
  if (t < 240) {  // row t of G_active: Gf[t,k], k = tid
    int k = tid;
    float gf = (k >= 16) ? ws[OFF_S + t * 240 + (k - 16)] : 0.f;
    float d = taut - ws[OFF_TAU + k] + OEPSc;
    float r = fmaxf(d, 0.f);
    rowg = gf * r;
    rowl = gf * r * r;
  }
  if (tid < 240 && t >= 16) {  // column t: Gf[i,t], i = tid
    float gf = ws[OFF_S + tid * 240 + (t - 16)];
    float d = ws[OFF_TAU + tid] - taut + OEPSc;
    colg = gf * fmaxf(d, 0.f);
  }
  r0[tid] = rowg; r1[tid] = rowl; r2[tid] = colg;
  __syncthreads();
  for (int s = 128; s > 0; s >>= 1) {
    if (tid < s) { r0[tid] += r0[tid + s]; r1[tid] += r1[tid + s]; r2[tid] += r2[tid + s]; }
    __syncthreads();
  }
  if (tid == 0) {
    ws[OFF_GT + t] = (r0[0] - r2[0]) * (1.f / 30720.f);   // 2/(240*256)
    if (r1[0] != 0.f) atomicAdd(ws + OFF_LOSS + 1, r1[0] * (1.f / 61440.f));
  }
}

__global__ void k_adam(float* ws, float rb1, float rb2) {
  if (((const int*)(ws + OFF_ACT))[0] == 0) return;
  int idx = blockIdx.x * blockDim.x + threadIdx.x;
  if (idx < 57600) {
    float g = ws[OFF_GG + idx];
    float m = B1c * ws[OFF_MG + idx] + (1.f - B1c) * g;
    float v = B2c * ws[OFF_VG + idx] + (1.f - B2c) * g * g;
    ws[OFF_MG + idx] = m; ws[OFF_VG + idx] = v;
    ws[OFF_GL + idx] -= LRc * (m * rb1) / (sqrtf(v * rb2) + EPSc);
  } else if (idx < 57856) {
    int j = idx - 57600;
    float g = ws[OFF_GT + j];
    float m = B1c * ws[OFF_MT + j] + (1.f - B1c) * g;
    float v = B2c * ws[OFF_VT + j] + (1.f - B2c) * g * g;
    ws[OFF_MT + j] = m; ws[OFF_VT + j] = v;
    ws[OFF_TAU + j] -= LRc * (m * rb1) / (sqrtf(v * rb2) + EPSc);
  }
}

// while_loop early-exit: checked strictly between body iterations.
__global__ void k_check(float* ws) {
  int* act = (int*)(ws + OFF_ACT);
  if (*act && (ws[OFF_LOSS] + ws[OFF_LOSS + 1] < TOLc)) *act = 0;
}

__global__ void k_final(const float* __restrict__ ws, float* __restrict__ out) {
  out[0] = ws[OFF_LOSS] + ws[OFF_LOSS + 1];
}

extern "C" void kernel_launch(void* const* d_in, const int* in_sizes, int n_in,
                              void* d_out, int out_size, void* d_ws, size_t ws_size,
                              hipStream_t stream) {
  const float* A    = (const float*)d_in[0];   // 256*256
  const float* Gl0  = (const float*)d_in[1];   // 240*240
  const float* tau0 = (const float*)d_in[2];   // 256
  float* ws  = (float*)d_ws;                   // needs ~1.39 MB
  float* out = (float*)d_out;
  (void)in_sizes; (void)n_in; (void)out_size; (void)ws_size;

  dim3 blk(256);
  dim3 gridE(225);        // 57600 / 256
  dim3 gridT(15, 15);     // 16x16 tiles over 240x240
  dim3 gridO(256);
  dim3 gridU(226);        // 57856 / 256

  k_init<<<gridE, blk, 0, stream>>>(Gl0, tau0, ws);

  float b1p = 1.f, b2p = 1.f;
  for (int it = 0; it < 30; ++it) {
    b1p *= B1c; b2p *= B2c;
    float rb1 = 1.f / (1.f - b1p);
    float rb2 = 1.f / (1.f - b2p);
    k_sigmoid<<<gridE, blk, 0, stream>>>(ws, 0);
    k_forward<<<gridT, blk, 0, stream>>>(A, ws, 0);
    k_gradG  <<<gridT, blk, 0, stream>>>(A, ws);
    k_order  <<<gridO, blk, 0, stream>>>(ws, 0);
    k_adam   <<<gridU, blk, 0, stream>>>(ws, rb1, rb2);
    k_check  <<<1, 1, 0, stream>>>(ws);
  }
  // Final loss at optimized params (always runs, matching the reference)
  k_sigmoid<<<gridE, blk, 0, stream>>>(ws, 1);
  k_forward<<<gridT, blk, 0, stream>>>(A, ws, 1);
  k_order  <<<gridO, blk, 0, stream>>>(ws, 1);
  k_final  <<<1, 1, 0, stream>>>(ws, out);
}